// LowRankModule_50268297232603
// MI455X (gfx1250) — compile-verified
//
#include <hip/hip_runtime.h>
#include <math.h>

typedef __attribute__((ext_vector_type(2))) float v2f;
typedef __attribute__((ext_vector_type(8))) float v8f;

constexpr int B = 2, H = 16, S = 2048, D = 64;
constexpr float SCALE = 0.125f;       // 1/sqrt(64)
constexpr int QT = 16;                // query rows per workgroup (one WMMA M tile)
constexpr int NKT = S / 16;           // 128 key tiles
constexpr int NWAVE = 8;              // 256 threads = 8 wave32
constexpr int SP = S + 4;             // padded LDS row stride (bank decorrelation)

__device__ __forceinline__ float waveMax(float v) {
#pragma unroll
  for (int off = 16; off > 0; off >>= 1) v = fmaxf(v, __shfl_xor(v, off, 32));
  return v;
}
__device__ __forceinline__ float waveSum(float v) {
#pragma unroll
  for (int off = 16; off > 0; off >>= 1) v += __shfl_xor(v, off, 32);
  return v;
}
__device__ __forceinline__ int waveSumI(int v) {
#pragma unroll
  for (int off = 16; off > 0; off >>= 1) v += __shfl_xor(v, off, 32);
  return v;
}

// One workgroup handles 16 query rows of one (b,h): WMMA QK^T into LDS,
// then per-row softmax + threshold-bisection top-k mass count.
__global__ void sparsity_kernel(const float* __restrict__ qg,
                                const float* __restrict__ kg,
                                float* __restrict__ out) {
  extern __shared__ float smem[];           // QT * SP floats (~128 KB)
  const int bh = blockIdx.x / (S / QT);
  const int q0 = (blockIdx.x % (S / QT)) * QT;
  const float* Q = qg + (size_t)bh * S * D;
  const float* K = kg + (size_t)bh * S * D;
  const int lane  = threadIdx.x & 31;
  const int wave  = threadIdx.x >> 5;
  const int m     = lane & 15;              // row (A) / col (B) within tile
  const int khalf = (lane >> 4) * 2;        // K sub-offset: 0 or 2

  // Preload 16x64 fp32 A fragments (16 k-steps of K=4 each)
  v2f a[16];
#pragma unroll
  for (int kk = 0; kk < 16; ++kk) {
    const int kc = kk * 4 + khalf;
    a[kk].x = Q[(q0 + m) * D + kc];
    a[kk].y = Q[(q0 + m) * D + kc + 1];
  }

  // Each wave strides over key tiles.
  for (int kt = wave; kt < NKT; kt += NWAVE) {
    // Prefetch next key tile into L2/L0 while this tile computes.
    if (kt + NWAVE < NKT)
      __builtin_prefetch(&K[((kt + NWAVE) * 16 + m) * D + khalf], 0, 1);

    // Stage all 16 B fragments first: 16 loads in flight, single wait.
    v2f b[16];
#pragma unroll
    for (int kk = 0; kk < 16; ++kk) {
      const int kc = kk * 4 + khalf;
      b[kk].x = K[(kt * 16 + m) * D + kc];
      b[kk].y = K[(kt * 16 + m) * D + kc + 1];
    }

    // Two accumulator chains for XDL ILP, merged at the end.
    v8f acc0 = {}, acc1 = {};
#pragma unroll
    for (int kk = 0; kk < 16; kk += 2) {
      acc0 = __builtin_amdgcn_wmma_f32_16x16x4_f32(
          false, a[kk],     false, b[kk],     (short)0, acc0, false, false);
      acc1 = __builtin_amdgcn_wmma_f32_16x16x4_f32(
          false, a[kk + 1], false, b[kk + 1], (short)0, acc1, false, false);
    }

    const int rbase = (lane >> 4) * 8;      // D layout: VGPR r -> row r or r+8
#pragma unroll
    for (int r = 0; r < 8; ++r)
      smem[(rbase + r) * SP + kt * 16 + m] = (acc0[r] + acc1[r]) * SCALE;
  }
  __syncthreads();

  // Per-row: softmax stats then bisection for minimal k with top-k mass >= 0.99.
  for (int row = wave; row < QT; row += NWAVE) {
    float* rp = smem + row * SP;
    float4* rp4 = reinterpret_cast<float4*>(rp);   // 512 float4 per row

    float mx = -3.4e38f;
    for (int j = lane; j < S / 4; j += 32) {
      const float4 v = rp4[j];
      mx = fmaxf(mx, fmaxf(fmaxf(v.x, v.y), fmaxf(v.z, v.w)));
    }
    mx = waveMax(mx);

    float se = 0.f;
    for (int j = lane; j < S / 4; j += 32) {
      float4 v = rp4[j];
      v.x = __expf(v.x - mx);
      v.y = __expf(v.y - mx);
      v.z = __expf(v.z - mx);
      v.w = __expf(v.w - mx);
      rp4[j] = v;                           // overwrite score with unnormalized prob
      se += v.x + v.y + v.z + v.w;
    }
    se = waveSum(se);
    const float target = 0.99f * se;

    // Largest t with G(t)=sum_{e>=t} e >= target; k* = count at that t.
    float lo = 0.f, hi = 1.0000002f;        // e in (0, 1]
    int cnt = S;
#pragma unroll 1
    for (int it = 0; it < 32; ++it) {
      const float mid = 0.5f * (lo + hi);
      float g = 0.f; int c = 0;
      for (int j = lane; j < S / 4; j += 32) {
        const float4 v = rp4[j];
        if (v.x >= mid) { g += v.x; ++c; }
        if (v.y >= mid) { g += v.y; ++c; }
        if (v.z >= mid) { g += v.z; ++c; }
        if (v.w >= mid) { g += v.w; ++c; }
      }
      g = waveSum(g);
      c = waveSumI(c);
      if (g >= target) { lo = mid; cnt = c; } else { hi = mid; }
    }
    if (lane == 0)
      out[(size_t)bh * S + q0 + row] = (float)(S - cnt + 1) / (float)S;
  }
}

__global__ void zero_ws_kernel(int* __restrict__ p, int n) {
  const int i = blockIdx.x * blockDim.x + threadIdx.x;
  if (i < n) p[i] = 0;
}

// Sparsity values are quantized to j/S -> exact percentiles via histograms.
__global__ void hist_kernel(const float* __restrict__ sp,
                            int* __restrict__ ghist, int* __restrict__ hhist) {
  const int idx = blockIdx.x * blockDim.x + threadIdx.x;
  if (idx >= B * H * S) return;
  const int h = (idx / S) % H;
  int v = (int)(sp[idx] * (float)S + 0.5f);
  v = v < 0 ? 0 : (v > S ? S : v);
  atomicAdd(&ghist[v], 1);
  atomicAdd(&hhist[h * (S + 1) + v], 1);
}

// block 0: global stats (n=B*H*S); block 1+h: per-head stats (n=B*S).
__global__ void stats_kernel(const int* __restrict__ ghist,
                             const int* __restrict__ hhist,
                             float* __restrict__ stats,
                             float* __restrict__ phstats) {
  const double P[9] = {0.001, 0.005, 0.01, 0.02, 0.05, 0.25, 0.5, 0.75, 0.95};
  const int t = threadIdx.x;
  if (t >= 10) return;
  const int* hist;
  int n;
  if (blockIdx.x == 0) { hist = ghist; n = B * H * S; }
  else { hist = hhist + (blockIdx.x - 1) * (S + 1); n = B * S; }
  int k;
  if (t == 0) k = 1;                        // sorted[0]
  else { const int kk = (int)(P[t - 1] * (double)n); k = kk < 1 ? 1 : kk; }
  int cum = 0, v = 0;
  for (v = 0; v <= S; ++v) { cum += hist[v]; if (cum >= k) break; }
  const float val = (float)v / (float)S;
  if (blockIdx.x == 0) stats[t] = val;
  else phstats[t * H + (int)(blockIdx.x - 1)] = val;
}

extern "C" void kernel_launch(void* const* d_in, const int* in_sizes, int n_in,
                              void* d_out, int out_size, void* d_ws, size_t ws_size,
                              hipStream_t stream) {
  (void)in_sizes; (void)n_in; (void)out_size; (void)ws_size;
  const float* q = (const float*)d_in[0];
  const float* k = (const float*)d_in[1];
  float* out = (float*)d_out;               // [B*H*S] + [10] + [10*H]
  int* ghist = (int*)d_ws;                  // 2049 bins
  int* hhist = ghist + (S + 1);             // 16 * 2049 bins

  const int nhist = (S + 1) * (H + 1);
  zero_ws_kernel<<<(nhist + 255) / 256, 256, 0, stream>>>(ghist, nhist);

  const dim3 grid(B * H * (S / QT));        // 4096 workgroups
  sparsity_kernel<<<grid, NWAVE * 32, QT * SP * sizeof(float), stream>>>(q, k, out);

  hist_kernel<<<(B * H * S + 255) / 256, 256, 0, stream>>>(out, ghist, hhist);

  stats_kernel<<<H + 1, 32, 0, stream>>>(ghist, hhist,
                                         out + (size_t)B * H * S,
                                         out + (size_t)B * H * S + 10);
}